// Criterion_35098472743267
// MI455X (gfx1250) — compile-verified
//
#include <hip/hip_runtime.h>
#include <math.h>

typedef __attribute__((ext_vector_type(2))) float v2f;
typedef __attribute__((ext_vector_type(8))) float v8f;

struct F3 { float x, y, z; };

__device__ __forceinline__ F3 f3_load(const float* __restrict__ base, int idx) {
  const float* p = base + (size_t)3 * (size_t)idx;
  F3 r; r.x = p[0]; r.y = p[1]; r.z = p[2]; return r;
}
__device__ __forceinline__ F3 f3_sub(F3 a, F3 b) { F3 r{a.x-b.x, a.y-b.y, a.z-b.z}; return r; }
__device__ __forceinline__ F3 f3_cross(F3 a, F3 b) {
  F3 r{a.y*b.z - a.z*b.y, a.z*b.x - a.x*b.z, a.x*b.y - a.y*b.x}; return r;
}
__device__ __forceinline__ float f3_dot(F3 a, F3 b) { return a.x*b.x + a.y*b.y + a.z*b.z; }
__device__ __forceinline__ F3 f3_normalize(F3 a) {
  float inv = rsqrtf(f3_dot(a, a));
  F3 r{a.x*inv, a.y*inv, a.z*inv}; return r;
}

// Faithful re-implementation of the reference _hinge_frame: returns rotated n1
// and rotated e (rotated n0 is unused by the loss).
__device__ __forceinline__ void hinge_frame(const float* __restrict__ pos,
                                            const int* __restrict__ faces,
                                            int fc0, int fc1, int ev0, int ev1,
                                            F3& n1_rot, F3& e_rot) {
  const int* fa = faces + (size_t)3 * (size_t)fc0;
  const int* fb = faces + (size_t)3 * (size_t)fc1;
  int a0 = fa[0], a1 = fa[1], a2 = fa[2];
  int b0 = fb[0], b1 = fb[1], b2 = fb[2];
  F3 pa0 = f3_load(pos, a0), pa1 = f3_load(pos, a1), pa2 = f3_load(pos, a2);
  F3 pb0 = f3_load(pos, b0), pb1 = f3_load(pos, b1), pb2 = f3_load(pos, b2);
  F3 n0 = f3_normalize(f3_cross(f3_sub(pa1, pa0), f3_sub(pa2, pa0)));
  F3 n1 = f3_normalize(f3_cross(f3_sub(pb1, pb0), f3_sub(pb2, pb0)));
  F3 e  = f3_sub(f3_load(pos, ev1), f3_load(pos, ev0));
  // rot rows: x = normalize(cross(y,z)), y = normalize(e), z = normalize(n0)
  F3 zh = f3_normalize(n0);
  F3 yh = f3_normalize(e);
  F3 xh = f3_normalize(f3_cross(yh, zh));
  n1_rot.x = f3_dot(xh, n1); n1_rot.y = f3_dot(yh, n1); n1_rot.z = f3_dot(zh, n1);
  e_rot.x  = f3_dot(xh, e);  e_rot.y  = f3_dot(yh, e);  e_rot.z  = f3_dot(zh, e);
}

// Wave32 all-lane sum using the CDNA5 matrix pipe.
// A (16x4 f32, v2f/lane): A[M][0] = p[M] (lanes 0-15 vgpr0), A[M][2] = p[M+16]
// (lanes 16-31 vgpr0), K=1,3 zeroed. B = ones(4x16). D[M][*] = p[M] + p[M+16].
// Summing the 8 D vgprs per-lane leaves Sum(rows 0..7) in lanes 0-15 and
// Sum(rows 8..15) in lanes 16-31; two shuffles finish the 32-lane total.
__device__ __forceinline__ float wave_sum32(float v) {
  v2f A; A.x = v;    A.y = 0.0f;
  v2f B; B.x = 1.0f; B.y = 1.0f;
  v8f C = {};
  C = __builtin_amdgcn_wmma_f32_16x16x4_f32(false, A, false, B, (short)0, C,
                                            false, false);
  float s = C[0] + C[1] + C[2] + C[3] + C[4] + C[5] + C[6] + C[7];
  return __shfl(s, 0, 32) + __shfl(s, 16, 32);
}

__global__ void __launch_bounds__(256)
bending_energy_kernel(const float* __restrict__ pred,
                      const float* __restrict__ rest,
                      const float* __restrict__ area,
                      const int* __restrict__ faces,
                      const int* __restrict__ fconn,
                      const int* __restrict__ fedge,
                      float* __restrict__ block_partials,
                      int num_edges) {
  float acc = 0.0f;
  const int stride = gridDim.x * blockDim.x;
  for (int i = blockIdx.x * blockDim.x + threadIdx.x; i < num_edges; i += stride) {
    int fc0 = fconn[2 * (size_t)i + 0];
    int fc1 = fconn[2 * (size_t)i + 1];
    int ev0 = fedge[2 * (size_t)i + 0];
    int ev1 = fedge[2 * (size_t)i + 1];

    F3 n1p, ep, n1q, eq;
    hinge_frame(pred, faces, fc0, fc1, ev0, ev1, n1p, ep);
    hinge_frame(rest, faces, fc0, fc1, ev0, ev1, n1q, eq);
    (void)eq;  // rest-frame rotated edge is unused by the loss

    float l2 = f3_dot(ep, ep);
    float inv_l = rsqrtf(l2);
    F3 en{ep.x * inv_l, ep.y * inv_l, ep.z * inv_l};

    float a = area[fc0] + area[fc1];
    float c = f3_dot(n1p, n1q);
    F3 cr = f3_cross(n1p, n1q);
    float s = f3_dot(en, cr);
    float theta = atan2f(s, c);

    acc += l2 / (4.0f * a) * theta * theta * 0.5f;
  }

  // Converged here: EXEC all-ones, legal for WMMA.
  float wave_total = wave_sum32(acc);

  __shared__ float lds[8];
  int lane = threadIdx.x & 31;
  int wid  = threadIdx.x >> 5;
  if (lane == 0) lds[wid] = wave_total;
  __syncthreads();
  if (threadIdx.x == 0) {
    float t = 0.0f;
#pragma unroll
    for (int w = 0; w < 8; ++w) t += lds[w];
    block_partials[blockIdx.x] = t;
  }
}

__global__ void __launch_bounds__(256)
final_reduce_kernel(const float* __restrict__ partials, int n,
                    const float* __restrict__ coeff,
                    float* __restrict__ out) {
  float acc = 0.0f;
  for (int i = threadIdx.x; i < n; i += blockDim.x) acc += partials[i];

  float wave_total = wave_sum32(acc);

  __shared__ float lds[8];
  int lane = threadIdx.x & 31;
  int wid  = threadIdx.x >> 5;
  if (lane == 0) lds[wid] = wave_total;
  __syncthreads();
  if (threadIdx.x == 0) {
    float t = 0.0f;
#pragma unroll
    for (int w = 0; w < 8; ++w) t += lds[w];
    out[0] = t * coeff[0];  // WEIGHT == 1.0
  }
}

extern "C" void kernel_launch(void* const* d_in, const int* in_sizes, int n_in,
                              void* d_out, int out_size, void* d_ws, size_t ws_size,
                              hipStream_t stream) {
  const float* pred  = (const float*)d_in[0];  // (V,3) f32
  const float* rest  = (const float*)d_in[1];  // (V,3) f32
  const float* area  = (const float*)d_in[2];  // (F,1) f32
  const float* coeff = (const float*)d_in[3];  // (1,)  f32
  const int*   faces = (const int*)d_in[4];    // (F,3) i32
  const int*   fconn = (const int*)d_in[5];    // (E,2) i32
  const int*   fedge = (const int*)d_in[6];    // (E,2) i32

  const int num_edges = in_sizes[5] / 2;
  const int NB = 1024;  // grid-stride; 1024*4B partials fit any workspace

  float* partials = (float*)d_ws;

  bending_energy_kernel<<<NB, 256, 0, stream>>>(pred, rest, area, faces, fconn,
                                                fedge, partials, num_edges);
  final_reduce_kernel<<<1, 256, 0, stream>>>(partials, NB, coeff, (float*)d_out);
}